// WassersteinLoss_78176994722580
// MI455X (gfx1250) — compile-verified
//
#include <hip/hip_runtime.h>

typedef float v2f __attribute__((ext_vector_type(2)));
typedef float v8f __attribute__((ext_vector_type(8)));

#define T_DIM 8192
#define NBLOCKS 2048
#define NTHREADS 256
#define STRIDE (NBLOCKS * NTHREADS)   // compile-time grid stride (vec4 units)

__device__ __forceinline__ float wacc(const float4 a, const float4 b, int vi, float acc) {
  const int   t  = (vi << 2) & (T_DIM - 1);   // row-local t (T pow2, vec4 never crosses row)
  const float w0 = (float)(T_DIM - t);        // weight[t] = T - t
  acc = fmaf(fabsf(a.x - b.x), w0,        acc);
  acc = fmaf(fabsf(a.y - b.y), w0 - 1.0f, acc);
  acc = fmaf(fabsf(a.z - b.z), w0 - 2.0f, acc);
  acc = fmaf(fabsf(a.w - b.w), w0 - 3.0f, acc);
  return acc;
}

// Pass 1: streaming weighted abs-diff, per-block partial sums.
__global__ __launch_bounds__(NTHREADS) void wloss_partial(
    const float* __restrict__ p, const float* __restrict__ q,
    float* __restrict__ partial, int n_vec) {
  const int tid = blockIdx.x * NTHREADS + threadIdx.x;
  const float4* __restrict__ pv = (const float4*)p;
  const float4* __restrict__ qv = (const float4*)q;

  float acc0 = 0.0f, acc1 = 0.0f, acc2 = 0.0f, acc3 = 0.0f;
  int vi = tid;
  // Unroll x4: 8 independent B128 loads in flight per trip, 4 FMA chains.
  for (; vi + 3 * STRIDE < n_vec; vi += 4 * STRIDE) {
    const float4 a0 = pv[vi + 0 * STRIDE], b0 = qv[vi + 0 * STRIDE];
    const float4 a1 = pv[vi + 1 * STRIDE], b1 = qv[vi + 1 * STRIDE];
    const float4 a2 = pv[vi + 2 * STRIDE], b2 = qv[vi + 2 * STRIDE];
    const float4 a3 = pv[vi + 3 * STRIDE], b3 = qv[vi + 3 * STRIDE];
    acc0 = wacc(a0, b0, vi + 0 * STRIDE, acc0);
    acc1 = wacc(a1, b1, vi + 1 * STRIDE, acc1);
    acc2 = wacc(a2, b2, vi + 2 * STRIDE, acc2);
    acc3 = wacc(a3, b3, vi + 3 * STRIDE, acc3);
  }
  for (; vi < n_vec; vi += STRIDE)    // remainder (empty when n_vec % (4*STRIDE) == 0)
    acc0 = wacc(pv[vi], qv[vi], vi, acc0);
  float acc = (acc0 + acc1) + (acc2 + acc3);

  // Wave32 cross-lane reduction via V_WMMA_F32_16X16X4_F32 with B = ones.
  // A(16x4): VGPR0 = acc (lanes 0-15 -> K=0, lanes 16-31 -> K=2), VGPR1 = 0.
  // => D[m][n] = acc[m] + acc[m+16].  Summing the 8 D VGPRs gives:
  //   lanes 0-15 : sum over M=0..7  of (acc[m]+acc[m+16])
  //   lanes 16-31: sum over M=8..15 of (acc[m]+acc[m+16])
  v2f A;  A[0] = acc;  A[1] = 0.0f;
  v2f B1; B1[0] = 1.0f; B1[1] = 1.0f;
  v8f C = {};
  C = __builtin_amdgcn_wmma_f32_16x16x4_f32(
      /*neg_a=*/false, A, /*neg_b=*/false, B1,
      /*c_mod=*/(short)0, C, /*reuse_a=*/false, /*reuse_b=*/false);
  const float s = ((C[0] + C[1]) + (C[2] + C[3])) + ((C[4] + C[5]) + (C[6] + C[7]));

  __shared__ float red[2 * (NTHREADS / 32)];
  const int lane = threadIdx.x & 31;
  const int wave = threadIdx.x >> 5;
  if (lane == 0)  red[2 * wave + 0] = s;   // holds acc[0..7] + acc[16..23]
  if (lane == 16) red[2 * wave + 1] = s;   // holds acc[8..15] + acc[24..31]
  __syncthreads();
  if (threadIdx.x == 0) {
    float blk = 0.0f;
    #pragma unroll
    for (int i = 0; i < 2 * (NTHREADS / 32); ++i) blk += red[i];
    partial[blockIdx.x] = blk;
  }
}

// Pass 2: single block, f64 accumulation of the per-block partials + scaling.
__global__ __launch_bounds__(NTHREADS) void wloss_final(
    const float* __restrict__ partial, int nblocks,
    float* __restrict__ out, double scale) {
  __shared__ double red[NTHREADS];
  double a = 0.0;
  for (int i = threadIdx.x; i < nblocks; i += NTHREADS) a += (double)partial[i];
  red[threadIdx.x] = a;
  __syncthreads();
  for (int s = NTHREADS / 2; s > 0; s >>= 1) {
    if (threadIdx.x < (unsigned)s) red[threadIdx.x] += red[threadIdx.x + s];
    __syncthreads();
  }
  if (threadIdx.x == 0) out[0] = (float)(red[0] * scale);
}

extern "C" void kernel_launch(void* const* d_in, const int* in_sizes, int n_in,
                              void* d_out, int out_size, void* d_ws, size_t ws_size,
                              hipStream_t stream) {
  const float* y_pred = (const float*)d_in[0];
  const float* y_true = (const float*)d_in[1];
  float* out     = (float*)d_out;
  float* partial = (float*)d_ws;           // NBLOCKS floats, fully rewritten every call

  const int N     = in_sizes[0];           // B*C*T = 16,777,216
  const int n_vec = N >> 2;

  // scaling * batch-mean:  2/(T*C*(T+1)) * (1/B) * total  =  2/((T+1)*N) * total
  const double scale = 2.0 / ((double)(T_DIM + 1) * (double)N);

  wloss_partial<<<NBLOCKS, NTHREADS, 0, stream>>>(y_pred, y_true, partial, n_vec);
  wloss_final<<<1, NTHREADS, 0, stream>>>(partial, NBLOCKS, out, scale);
}